// RestrictedConvolutionalBlock_83803401880090
// MI455X (gfx1250) — compile-verified
//
#include <hip/hip_runtime.h>

typedef __attribute__((ext_vector_type(2))) float v2f;
typedef __attribute__((ext_vector_type(8))) float v8f;

#define N_J    53
#define C_IN   32
#define C_OUT  32
#define KW     3
#define BB     16
#define TLEN   2048
#define CHUNK  256
#define TILES_H (CHUNK/16 + 2)   /* 18: 16 real tiles + 2 halo tiles */
#define TILES_Y (CHUNK/16)       /* 16 */
#define HROWS  (CHUNK + 32)      /* 288 rows of h in LDS */
#define HSTRIDE 34               /* 32 + 2 pad: lane halves hit disjoint banks */
#define PSTR   96                /* pair-stride for repacked weights: 64 + 32 pad */

__device__ __constant__ int NAME_IDS_D[N_J] = {
    0, 1, 2, 3, 4, 1, 2, 3, 4, 5, 6, 7, 8, 9, 10, 11, 12, 13, 14, 15, 16,
    14, 15, 16, 14, 15, 16, 14, 15, 16, 14, 15, 16, 10, 11, 12, 13, 14, 15,
    16, 14, 15, 16, 14, 15, 16, 14, 15, 16, 14, 15, 16, 17};

__global__ __launch_bounds__(256)
void fused_skel_block(const float* __restrict__ x,
                      const float* __restrict__ lin_w,
                      const float* __restrict__ lin_b,
                      const float* __restrict__ conv_w,
                      const float* __restrict__ conv_b,
                      float* __restrict__ out)
{
    // hbuf: 288*34 | lwp: 32 f-pairs * 96 | cwp: 3k * 16 c-pairs * 96 | biases
    __shared__ float smem[HROWS * HSTRIDE + 32 * PSTR + KW * 16 * PSTR + 64];
    float* hbuf = smem;
    float* lwp  = smem + HROWS * HSTRIDE;       // [f/2][2*o+q] q=f&1, stride 96
    float* cwp  = lwp + 32 * PSTR;              // [k][c/2][2*o+q] q=c&1
    float* lb   = cwp + KW * 16 * PSTR;         // 32
    float* cb   = lb + C_OUT;                   // 32

    const int tid  = threadIdx.x;
    const int lane = tid & 31;
    const int wave = tid >> 5;
    const int m    = lane & 15;          // WMMA row (M) this lane owns in A
    const int half = lane >> 4;          // 0: K-slots 0,1   1: K-slots 2,3
    const int kb   = half << 1;
    const int o0   = lane & 15;          // WMMA col (N) this lane owns in B/D

    const int chunk = blockIdx.x;
    const int j     = blockIdx.y;
    const int b     = blockIdx.z;
    const int t0    = chunk * CHUNK;
    const int jp    = (j > 0) ? (j - 1) : 0;
    const int name  = NAME_IDS_D[j];

    // ---- preload per-joint weights into LDS, repacked for b64 B-fragments ----
    {
        const float* s = lin_w + (size_t)name * (2 * C_IN * C_OUT);
        for (int idx = tid; idx < 2 * C_IN * C_OUT; idx += 256) {
            int f = idx >> 5;            // 0..63
            int o = idx & 31;
            lwp[(f >> 1) * PSTR + 2 * o + (f & 1)] = s[idx];
        }
        const float* s2 = conv_w + (size_t)name * (C_OUT * C_OUT * KW);
        for (int idx = tid; idx < C_OUT * C_OUT * KW; idx += 256) {
            int o = idx / (C_OUT * KW);
            int r = idx - o * (C_OUT * KW);
            int c = r / KW;
            int k = r - c * KW;
            cwp[(k * 16 + (c >> 1)) * PSTR + 2 * o + (c & 1)] = s2[idx];
        }
        if (tid < C_OUT) {
            lb[tid] = lin_b[name * C_OUT + tid];
            cb[tid] = conv_b[name * C_OUT + tid];
        }
    }
    __syncthreads();

    // ---- stage 1: gather(own,parent) -> 64x32 linear -> ReLU -> LDS h ----
    for (int i = wave; i < TILES_H; i += 8) {
        const int ttile = t0 - 16 + i * 16;  // includes halo tiles
        int t  = ttile + m;
        int tc = t < 0 ? 0 : (t > TLEN - 1 ? TLEN - 1 : t);  // clamp: halo-only
        const float* rowOwn = x + ((size_t)((size_t)b * TLEN + tc) * N_J + j)  * C_IN;
        const float* rowPar = x + ((size_t)((size_t)b * TLEN + tc) * N_J + jp) * C_IN;

        // prefetch this wave's next tile rows (global_prefetch_b8)
        {
            int tn  = ttile + 8 * 16 + m;
            int tcn = tn < 0 ? 0 : (tn > TLEN - 1 ? TLEN - 1 : tn);
            const float* pOwn = x + ((size_t)((size_t)b * TLEN + tcn) * N_J + j)  * C_IN;
            const float* pPar = x + ((size_t)((size_t)b * TLEN + tcn) * N_J + jp) * C_IN;
            __builtin_prefetch(pOwn, 0, 1);
            __builtin_prefetch(pPar, 0, 1);
        }

        v8f acc0 = {};
        v8f acc1 = {};
        #pragma unroll
        for (int k0 = 0; k0 < 2 * C_IN; k0 += 4) {
            const int f = k0 + kb;
            const float* src = (f < C_IN) ? (rowOwn + f) : (rowPar + (f - C_IN));
            v2f a = *(const v2f*)src;                      // A[m,f], A[m,f+1]
            const float* wp = lwp + ((k0 >> 1) + half) * PSTR;
            v2f bf0 = *(const v2f*)(wp + 2 * o0);          // W[f][o0], W[f+1][o0]
            v2f bf1 = *(const v2f*)(wp + 2 * (16 + o0));
            acc0 = __builtin_amdgcn_wmma_f32_16x16x4_f32(false, a, false, bf0,
                                                         (short)0, acc0, false, false);
            acc1 = __builtin_amdgcn_wmma_f32_16x16x4_f32(false, a, false, bf1,
                                                         (short)0, acc1, false, false);
        }
        const int mbase = half * 8;          // D rows: r (lanes 0-15), r+8 (16-31)
        #pragma unroll
        for (int r = 0; r < 8; ++r) {
            const int row = i * 16 + mbase + r;
            float v0 = acc0[r] + lb[o0];
            float v1 = acc1[r] + lb[16 + o0];
            hbuf[row * HSTRIDE + o0]      = fmaxf(v0, 0.0f);
            hbuf[row * HSTRIDE + 16 + o0] = fmaxf(v1, 0.0f);
        }
    }
    __syncthreads();

    // ---- stage 2: K=3 reflect-padded grouped conv as 16x32x(3*32) WMMA ----
    for (int i = wave; i < TILES_Y; i += 8) {
        const int ttile = t0 + i * 16;
        v8f acc0 = {};
        v8f acc1 = {};
        #pragma unroll
        for (int k = 0; k < KW; ++k) {
            int t = ttile + m - 1 + k;
            if (t < 0)        t = -t;                  // reflect: -1 -> 1
            if (t > TLEN - 1) t = 2 * (TLEN - 1) - t;  // reflect: T -> T-2
            const float* hrow = hbuf + (t - (t0 - 16)) * HSTRIDE;
            #pragma unroll
            for (int c0s = 0; c0s < C_OUT; c0s += 4) {
                const int cc = c0s + kb;
                v2f a = *(const v2f*)(hrow + cc);      // h[t,cc], h[t,cc+1]
                const float* wp = cwp + (k * 16 + (c0s >> 1) + half) * PSTR;
                v2f bf0 = *(const v2f*)(wp + 2 * o0);  // cw[k][cc][o0], [cc+1][o0]
                v2f bf1 = *(const v2f*)(wp + 2 * (16 + o0));
                acc0 = __builtin_amdgcn_wmma_f32_16x16x4_f32(false, a, false, bf0,
                                                             (short)0, acc0, false, false);
                acc1 = __builtin_amdgcn_wmma_f32_16x16x4_f32(false, a, false, bf1,
                                                             (short)0, acc1, false, false);
            }
        }
        const int mbase = half * 8;
        #pragma unroll
        for (int r = 0; r < 8; ++r) {
            const int t = ttile + mbase + r;
            float* orow = out + ((size_t)((size_t)b * TLEN + t) * N_J + j) * C_OUT;
            orow[o0]      = acc0[r] + cb[o0];
            orow[16 + o0] = acc1[r] + cb[16 + o0];
        }
    }
}

extern "C" void kernel_launch(void* const* d_in, const int* in_sizes, int n_in,
                              void* d_out, int out_size, void* d_ws, size_t ws_size,
                              hipStream_t stream) {
    const float* x      = (const float*)d_in[0];
    const float* lin_w  = (const float*)d_in[1];
    const float* lin_b  = (const float*)d_in[2];
    const float* conv_w = (const float*)d_in[3];
    const float* conv_b = (const float*)d_in[4];
    float* out = (float*)d_out;

    dim3 grid(TLEN / CHUNK, N_J, BB);   // 8 x 53 x 16 workgroups
    fused_skel_block<<<grid, 256, 0, stream>>>(x, lin_w, lin_b, conv_w, conv_b, out);
}